// dcn_56925496541545
// MI455X (gfx1250) — compile-verified
//
#include <hip/hip_runtime.h>

typedef __attribute__((ext_vector_type(2))) float v2f;
typedef __attribute__((ext_vector_type(8))) float v8f;

#define T3 (1.0f/3.0f)
#define F5 (0.2f)
#define E8 (0.125f)

// S padded to 16x12 (WMMA A-matrix = stencil). Rows 9-15 and cols 9-11 zero.
__device__ __constant__ float c_S[16][12] = {
    {0.f, T3,  0.f, T3,  T3,  0.f, 0.f, 0.f, 0.f, 0.f, 0.f, 0.f}, // b00
    {F5,  0.f, F5,  F5,  F5,  F5,  0.f, 0.f, 0.f, 0.f, 0.f, 0.f}, // b01
    {0.f, T3,  0.f, 0.f, T3,  T3,  0.f, 0.f, 0.f, 0.f, 0.f, 0.f}, // b02
    {F5,  F5,  0.f, 0.f, F5,  0.f, F5,  F5,  0.f, 0.f, 0.f, 0.f}, // b10
    {E8,  E8,  E8,  E8,  0.f, E8,  E8,  E8,  E8,  0.f, 0.f, 0.f}, // b11
    {0.f, F5,  F5,  0.f, F5,  0.f, 0.f, F5,  F5,  0.f, 0.f, 0.f}, // b12
    {0.f, 0.f, 0.f, T3,  T3,  0.f, 0.f, T3,  0.f, 0.f, 0.f, 0.f}, // b20
    {0.f, 0.f, 0.f, F5,  F5,  F5,  F5,  0.f, F5,  0.f, 0.f, 0.f}, // b21
    {0.f, 0.f, 0.f, 0.f, T3,  T3,  0.f, T3,  0.f, 0.f, 0.f, 0.f}, // b22
    {0.f,0.f,0.f,0.f,0.f,0.f,0.f,0.f,0.f,0.f,0.f,0.f},
    {0.f,0.f,0.f,0.f,0.f,0.f,0.f,0.f,0.f,0.f,0.f,0.f},
    {0.f,0.f,0.f,0.f,0.f,0.f,0.f,0.f,0.f,0.f,0.f,0.f},
    {0.f,0.f,0.f,0.f,0.f,0.f,0.f,0.f,0.f,0.f,0.f,0.f},
    {0.f,0.f,0.f,0.f,0.f,0.f,0.f,0.f,0.f,0.f,0.f,0.f},
    {0.f,0.f,0.f,0.f,0.f,0.f,0.f,0.f,0.f,0.f,0.f,0.f},
    {0.f,0.f,0.f,0.f,0.f,0.f,0.f,0.f,0.f,0.f,0.f,0.f},
};

// Swap lane halves (lane ^= 16): ds_swizzle group-of-32, xor=0x10, or=0, and=0x1f.
__device__ __forceinline__ float swap16(float v) {
    return __int_as_float(__builtin_amdgcn_ds_swizzle(__float_as_int(v), 0x401f));
}

// sigmoid(z) = rcp(1 + exp2(-z*log2(e))) -> v_exp_f32 + v_rcp_f32
__device__ __forceinline__ float sigm(float z) {
    float e = __builtin_amdgcn_exp2f(z * -1.44269504088896340736f);
    return __builtin_amdgcn_rcpf(1.0f + e);
}

__global__ void __launch_bounds__(256)
stencil3_sigmoid_wmma(const float* __restrict__ xin,
                      const float* __restrict__ w1p,
                      float* __restrict__ out,
                      int n, long long tiles) {
    const int  lane = threadIdx.x & 31;
    const int  m    = lane & 15;
    const bool hiH  = lane >= 16;

    const float w1 = w1p[0];
    const float w2 = 1.0f - w1;

    // A = S fragments in WMMA 16x4 f32 A-layout:
    // lane m holds K = k0, k0+1 ; lane m+16 holds K = k0+2, k0+3 for each chunk.
    const int k0 = hiH ? 2 : 0;
    v2f a0, a1, a2;
    a0[0] = c_S[m][0 + k0]; a0[1] = c_S[m][1 + k0];
    a1[0] = c_S[m][4 + k0]; a1[1] = c_S[m][5 + k0];
    a2[0] = c_S[m][8 + k0]; a2[1] = c_S[m][9 + k0];

    const long long nw = (long long)gridDim.x * (blockDim.x >> 5);
    long long gw = (long long)blockIdx.x * (blockDim.x >> 5) + (threadIdx.x >> 5);

    for (long long t = gw; t < tiles; t += nw) {
        long long s  = t * 16 + m;                        // this lane's sample column
        long long sc = (s < n) ? s : (long long)n - 1;    // clamp (N%16==0 normally)
        const float* p = xin + sc * 9;

        // x held in WMMA D-layout: lane n (lo) holds elements 0..7 of sample n in
        // x0..x7; lane n+16 holds element 8 in x0, padding elsewhere.
        float x0, x1, x2, x3, x4, x5, x6, x7;
        if (!hiH) {
            x0 = p[0]; x1 = p[1]; x2 = p[2]; x3 = p[3];
            x4 = p[4]; x5 = p[5]; x6 = p[6]; x7 = p[7];
        } else {
            x0 = p[8];
            x1 = x2 = x3 = x4 = x5 = x6 = x7 = 0.0f;
        }

        // Two full stencil+sigmoid iterations via WMMA (K = 0..11, 3 chunks of 4).
#pragma unroll
        for (int it = 0; it < 2; ++it) {
            // Rebuild B (element x sample) from D-layout x: 5 half-swaps + selects.
            float s2 = swap16(x2);   // element 2 -> hi lanes
            float s3 = swap16(x3);   // element 3 -> hi lanes
            float s6 = swap16(x6);   // element 6 -> hi lanes
            float s7 = swap16(x7);   // element 7 -> hi lanes
            float s0 = swap16(x0);   // element 8 (lives in hi x0) -> lo lanes

            v2f b0, b1, b2;
            b0[0] = hiH ? s2 : x0;   b0[1] = hiH ? s3 : x1;   // rows 0,2 | 1,3
            b1[0] = hiH ? s6 : x4;   b1[1] = hiH ? s7 : x5;   // rows 4,6 | 5,7
            b2[0] = hiH ? 0.0f : s0; b2[1] = 0.0f;            // rows 8,10 | 9,11

            v8f acc = {0.f,0.f,0.f,0.f,0.f,0.f,0.f,0.f};
            acc = __builtin_amdgcn_wmma_f32_16x16x4_f32(false, a0, false, b0,
                                                        (short)0, acc, false, false);
            acc = __builtin_amdgcn_wmma_f32_16x16x4_f32(false, a1, false, b1,
                                                        (short)0, acc, false, false);
            acc = __builtin_amdgcn_wmma_f32_16x16x4_f32(false, a2, false, b2,
                                                        (short)0, acc, false, false);

            // D-layout matches x-layout: elementwise, fully lane-local.
            x0 = sigm(w1 * x0 + w2 * acc[0]);
            x1 = sigm(w1 * x1 + w2 * acc[1]);
            x2 = sigm(w1 * x2 + w2 * acc[2]);
            x3 = sigm(w1 * x3 + w2 * acc[3]);
            x4 = sigm(w1 * x4 + w2 * acc[4]);
            x5 = sigm(w1 * x5 + w2 * acc[5]);
            x6 = sigm(w1 * x6 + w2 * acc[6]);
            x7 = sigm(w1 * x7 + w2 * acc[7]);
        }

        // Iteration 3: only element 0 survives -> b00 = (a01+a10+a11)/3, lane-local.
        float nbr0 = (x1 + x3 + x4) * T3;
        float r = sigm(w1 * x0 + w2 * nbr0);
        if (!hiH && s < (long long)n) out[s] = r;
    }
}

extern "C" void kernel_launch(void* const* d_in, const int* in_sizes, int n_in,
                              void* d_out, int out_size, void* d_ws, size_t ws_size,
                              hipStream_t stream) {
    (void)in_sizes; (void)n_in; (void)d_ws; (void)ws_size;
    const float* x  = (const float*)d_in[0];
    const float* w1 = (const float*)d_in[1];
    float* out = (float*)d_out;

    const int n = out_size;                      // N samples
    const long long tiles = ((long long)n + 15) / 16;

    const int threads = 256;                     // 8 waves per block (wave32)
    const int wavesPerBlock = threads / 32;
    long long blocksNeeded = (tiles + wavesPerBlock - 1) / wavesPerBlock;
    int blocks = (int)(blocksNeeded < 8192 ? blocksNeeded : 8192);
    if (blocks < 1) blocks = 1;

    stencil3_sigmoid_wmma<<<blocks, threads, 0, stream>>>(x, w1, out, n, tiles);
}